// GCN_27212912787870
// MI455X (gfx1250) — compile-verified
//
#include <hip/hip_runtime.h>

#define FDIM 128

typedef __attribute__((ext_vector_type(2))) float v2f;
typedef __attribute__((ext_vector_type(8))) float v8f;

// ---------------------------------------------------------------------------
// Degree / normalization kernels
// ---------------------------------------------------------------------------
__global__ __launch_bounds__(256) void k_deg_init(float* __restrict__ deg, int n) {
    int i = blockIdx.x * blockDim.x + threadIdx.x;
    if (i < n) deg[i] = 1.0f;  // self-loop contributes 1 to every node's degree
}

__global__ __launch_bounds__(256) void k_deg_acc(const int* __restrict__ dst,
                                                 float* __restrict__ deg, int e) {
    int i = blockIdx.x * blockDim.x + threadIdx.x;
    if (i < e) atomicAdd(&deg[dst[i]], 1.0f);
}

__global__ __launch_bounds__(256) void k_dinv(const float* __restrict__ deg,
                                              float* __restrict__ dinv, int n) {
    int i = blockIdx.x * blockDim.x + threadIdx.x;
    if (i < n) dinv[i] = rsqrtf(deg[i]);  // deg >= 1 always
}

// ---------------------------------------------------------------------------
// fp32 GEMM via V_WMMA_F32_16X16X4_F32.
// One wave computes a 16-row x 128-col output stripe (8 WMMA tiles).
// A: [rows x 128] row-major, B: [128 x 128] row-major, C: [rows x 128].
// RELU applies max(a,0) to A elements (fused ReLU of previous layer).
// Assumes rows % 16 == 0 (N = 100000 -> 6250 row tiles).
// ---------------------------------------------------------------------------
template <bool RELU>
__global__ __launch_bounds__(256) void k_gemm128(const float* __restrict__ A,
                                                 const float* __restrict__ B,
                                                 float* __restrict__ C,
                                                 int rowTiles) {
    const int wave = (int)((blockIdx.x * blockDim.x + threadIdx.x) >> 5);
    const int lane = (int)(threadIdx.x & 31);
    if (wave >= rowTiles) return;  // wave-uniform: EXEC all-1s inside

    const int half = lane >> 4;   // 0: K pair {0,1}; 1: K pair {2,3}
    const int lid  = lane & 15;   // row within A tile / col within B tile

    const float* arow = A + (size_t)(wave * 16 + lid) * FDIM;

    v8f acc[8];
#pragma unroll
    for (int t = 0; t < 8; ++t)
        acc[t] = v8f{0.f, 0.f, 0.f, 0.f, 0.f, 0.f, 0.f, 0.f};

#pragma unroll 4
    for (int k0 = 0; k0 < FDIM; k0 += 4) {
        const int ka = k0 + half * 2;
        float a0 = arow[ka];
        float a1 = arow[ka + 1];
        if (RELU) {
            a0 = fmaxf(a0, 0.f);
            a1 = fmaxf(a1, 0.f);
        }
        v2f av;
        av[0] = a0;
        av[1] = a1;
        const float* brow0 = B + (size_t)ka * FDIM + lid;        // row ka
        const float* brow1 = brow0 + FDIM;                       // row ka+1
#pragma unroll
        for (int t = 0; t < 8; ++t) {
            v2f bv;
            bv[0] = brow0[t * 16];
            bv[1] = brow1[t * 16];
            acc[t] = __builtin_amdgcn_wmma_f32_16x16x4_f32(
                false, av, false, bv, (short)0, acc[t], false, false);
        }
    }

    // C/D layout: VGPR v, lane half h, lane lid -> row = 16*wave + 8*h + v, col = lid
    float* cbase = C + (size_t)(wave * 16 + half * 8) * FDIM + lid;
#pragma unroll
    for (int t = 0; t < 8; ++t) {
#pragma unroll
        for (int v = 0; v < 8; ++v) {
            cbase[(size_t)v * FDIM + t * 16] = acc[t][v];
        }
    }
}

// ---------------------------------------------------------------------------
// agg[i,:] = bias[:] + dinv[i]^2 * h[i,:]   (bias + self-loop term)
// One thread per float4 -> N*32 threads. Fully initializes agg.
// ---------------------------------------------------------------------------
__global__ __launch_bounds__(256) void k_init_agg(const float* __restrict__ h,
                                                  const float* __restrict__ bias,
                                                  const float* __restrict__ dinv,
                                                  float* __restrict__ agg, int n) {
    int idx = blockIdx.x * blockDim.x + threadIdx.x;
    int i = idx >> 5;
    int q = idx & 31;
    if (i >= n) return;
    float di = dinv[i];
    float self = di * di;
    float4 hv = ((const float4*)(h + (size_t)i * FDIM))[q];
    float4 bv = ((const float4*)bias)[q];
    float4 o;
    o.x = bv.x + self * hv.x;
    o.y = bv.y + self * hv.y;
    o.z = bv.z + self * hv.z;
    o.w = bv.w + self * hv.w;
    ((float4*)(agg + (size_t)i * FDIM))[q] = o;
}

// ---------------------------------------------------------------------------
// Edge scatter: one wave per edge; lane l handles floats [4l, 4l+4).
// agg[dst,:] += dinv[src]*dinv[dst] * h[src,:]  via f32 atomics (hit in L2).
// ---------------------------------------------------------------------------
__global__ __launch_bounds__(256) void k_scatter(const float* __restrict__ h,
                                                 const int* __restrict__ src,
                                                 const int* __restrict__ dst,
                                                 const float* __restrict__ dinv,
                                                 float* __restrict__ agg, int e) {
    int wave = (int)((blockIdx.x * blockDim.x + threadIdx.x) >> 5);
    int lane = (int)(threadIdx.x & 31);
    if (wave >= e) return;
    int s = src[wave];
    int d = dst[wave];
    float nrm = dinv[s] * dinv[d];
    float4 v = ((const float4*)(h + (size_t)s * FDIM))[lane];
    float* out = agg + (size_t)d * FDIM + lane * 4;
    atomicAdd(out + 0, nrm * v.x);
    atomicAdd(out + 1, nrm * v.y);
    atomicAdd(out + 2, nrm * v.z);
    atomicAdd(out + 3, nrm * v.w);
}

// ---------------------------------------------------------------------------
extern "C" void kernel_launch(void* const* d_in, const int* in_sizes, int n_in,
                              void* d_out, int out_size, void* d_ws, size_t ws_size,
                              hipStream_t stream) {
    const float* x  = (const float*)d_in[0];
    const int*   ei = (const int*)d_in[1];
    const float* W1 = (const float*)d_in[2];
    const float* b1 = (const float*)d_in[3];
    const float* W2 = (const float*)d_in[4];
    const float* b2 = (const float*)d_in[5];
    (void)n_in; (void)out_size; (void)ws_size;

    const int N = in_sizes[0] / FDIM;
    const int E = in_sizes[1] / 2;
    const int* srcp = ei;       // edge_index[0]
    const int* dstp = ei + E;   // edge_index[1]
    float* out = (float*)d_out;

    // Workspace layout: deg[N] | dinv[N] | hbuf[N*128]  (~52 MB)
    float* deg  = (float*)d_ws;
    float* dinv = deg + N;
    float* hbuf = dinv + N;

    const int tpb = 256;
    const int rowTiles = N / 16;                 // N = 100000 -> 6250
    const int gemmBlocks = (rowTiles * 32 + tpb - 1) / tpb;
    const int initBlocks = (N * 32 + tpb - 1) / tpb;
    const long long scatterThreads = (long long)E * 32;
    const int scatterBlocks = (int)((scatterThreads + tpb - 1) / tpb);

    // Normalization: deg (with self-loops) -> dinv
    k_deg_init<<<(N + tpb - 1) / tpb, tpb, 0, stream>>>(deg, N);
    k_deg_acc<<<(E + tpb - 1) / tpb, tpb, 0, stream>>>(dstp, deg, E);
    k_dinv<<<(N + tpb - 1) / tpb, tpb, 0, stream>>>(deg, dinv, N);

    // ---- Layer 1: h1 = x @ W1; agg1 (in d_out) = b1 + D^-1/2 A D^-1/2 h1
    k_gemm128<false><<<gemmBlocks, tpb, 0, stream>>>(x, W1, hbuf, rowTiles);
    k_init_agg<<<initBlocks, tpb, 0, stream>>>(hbuf, b1, dinv, out, N);
    k_scatter<<<scatterBlocks, tpb, 0, stream>>>(hbuf, srcp, dstp, dinv, out, E);

    // ---- Layer 2: h2 = relu(agg1) @ W2 (ReLU fused into A-load); out = b2 + aggregate
    k_gemm128<true><<<gemmBlocks, tpb, 0, stream>>>(out, W2, hbuf, rowTiles);
    k_init_agg<<<initBlocks, tpb, 0, stream>>>(hbuf, b2, dinv, out, N);
    k_scatter<<<scatterBlocks, tpb, 0, stream>>>(hbuf, srcp, dstp, dinv, out, E);
}